// _LinearPairwiseModule_26714696581275
// MI455X (gfx1250) — compile-verified
//
#include <hip/hip_runtime.h>

// ---------------------------------------------------------------------------
// Problem: out[b] = sum_{d,v} x[b,d,v]*h[d,v]
//                 + sum_p x[b,pair_i[p],:]^T J[p] x[b,pair_j[p],:]  + bias
// B=32768, D=64, V=21, P=2016, NTOT = D*V = 1344.
//
// Reformulated as a dense fp32 GEMM + fused row-dot:
//   M[u,t] = J[p(i,j), v, w]  for u=(i,v), t=(j,w), i<j, else 0   (1344x1344)
//   out[b] = sum_t x[b,t] * ( (x[b,:] @ M)[:,t] + h[t] ) + bias
// Executed with V_WMMA_F32_16X16X4_F32 (full fp32 precision, shapes divide
// evenly: 1344 = 84*16 = 336*4). M (7.2MB) lives in L2; x streamed once.
// ---------------------------------------------------------------------------

typedef __attribute__((ext_vector_type(2))) float v2f;
typedef __attribute__((ext_vector_type(8))) float v8f;

#define NTOT 1344      // D*V
#define VDIM 21
#define NT_TILES 84    // NTOT / 16
#define KSTEPS 336     // NTOT / 4
#define WAVES 8        // 256 threads per block
#define BTILE 16       // batches per block

// ---------------------------------------------------------------------------
// Kernel 1: scatter J (P,V,V) into dense block-upper-triangular M (NTOT,NTOT)
// ---------------------------------------------------------------------------
__global__ void build_M_kernel(const float* __restrict__ J, float* __restrict__ M) {
    int idx = blockIdx.x * blockDim.x + threadIdx.x;
    if (idx >= NTOT * NTOT) return;
    int u = idx / NTOT;
    int t = idx - u * NTOT;
    int i = u / VDIM, v = u - i * VDIM;
    int j = t / VDIM, w = t - j * VDIM;
    float val = 0.0f;
    if (i < j) {
        // pair index for row-major (i<j) enumeration, D=64
        int p = i * 63 - (i * (i - 1)) / 2 + (j - i - 1);
        val = J[p * (VDIM * VDIM) + v * VDIM + w];
    }
    M[idx] = val;
}

// ---------------------------------------------------------------------------
// Kernel 2: fused  Z = X @ M  (WMMA f32 16x16x4)  with epilogue
//           out[b] = sum_t x[b,t]*(Z[b,t]+h[t]) + bias
// One block = 16 batches, 8 waves; each wave owns N-tiles nt = wave, wave+8,...
// ---------------------------------------------------------------------------
__global__ void __launch_bounds__(256)
fused_quadform_kernel(const float* __restrict__ x,
                      const float* __restrict__ h,
                      const float* __restrict__ M,
                      const float* __restrict__ bias,
                      float* __restrict__ out) {
    __shared__ float acc[BTILE];

    const int tid  = threadIdx.x;
    const int lane = tid & 31;
    const int wave = tid >> 5;
    const int col  = lane & 15;          // N-column within tile; also A-row (batch)
    const int hi   = (lane >= 16) ? 1 : 0;

    if (tid < BTILE) acc[tid] = 0.0f;
    __syncthreads();

    const int bbase = blockIdx.x * BTILE;

    // A-fragment source row: batch = bbase + (lane%16); K-slot offset 0 or 2.
    const float* xrow = x + (size_t)(bbase + col) * NTOT;
    const int koff = hi * 2;

    float rowacc[8];
#pragma unroll
    for (int r = 0; r < 8; ++r) rowacc[r] = 0.0f;

    for (int nt = wave; nt < NT_TILES; nt += WAVES) {
        v8f C = {0.f, 0.f, 0.f, 0.f, 0.f, 0.f, 0.f, 0.f};
        const int ncol = nt * 16 + col;
        const float* Mcol = M + ncol;    // element (k, ncol) at Mcol[k*NTOT]

        for (int k = 0; k < NTOT; k += 4) {
            const int kb = k + koff;
            // A 16x4 f32: VGPR0 = K=kb, VGPR1 = K=kb+1 (per-lane contiguous pair)
            v2f a = *(const v2f*)(xrow + kb);
            // B 4x16 f32: VGPR0 = row kb, VGPR1 = row kb+1, col = lane%16
            v2f b;
            b.x = Mcol[(size_t)kb * NTOT];
            b.y = Mcol[(size_t)(kb + 1) * NTOT];
            C = __builtin_amdgcn_wmma_f32_16x16x4_f32(
                    /*neg_a=*/false, a, /*neg_b=*/false, b,
                    /*c_mod=*/(short)0, C, /*reuse_a=*/false, /*reuse_b=*/false);
        }

        // Epilogue: rowacc[r] += (Z + h[t]) * x[brow, t] for this tile's column
        const int t = nt * 16 + col;
        const float hv = h[t];
#pragma unroll
        for (int r = 0; r < 8; ++r) {
            const int brow = r + hi * 8;   // C layout: row = r + (lane>=16 ? 8 : 0)
            const float xv = x[(size_t)(bbase + brow) * NTOT + t];
            rowacc[r] += (C[r] + hv) * xv;
        }
    }

    // Reduce across the 16 lanes that hold the 16 columns of each row
    // (xor offsets < 16 stay within each half-wave group).
#pragma unroll
    for (int r = 0; r < 8; ++r) {
        float v = rowacc[r];
        v += __shfl_xor(v, 1);
        v += __shfl_xor(v, 2);
        v += __shfl_xor(v, 4);
        v += __shfl_xor(v, 8);
        if (col == 0) {
            atomicAdd(&acc[r + hi * 8], v);
        }
    }
    __syncthreads();

    if (tid < BTILE) {
        out[bbase + tid] = acc[tid] + bias[0];
    }
}

// ---------------------------------------------------------------------------
extern "C" void kernel_launch(void* const* d_in, const int* in_sizes, int n_in,
                              void* d_out, int out_size, void* d_ws, size_t ws_size,
                              hipStream_t stream) {
    const float* x    = (const float*)d_in[0];  // (B, D, V)
    const float* h    = (const float*)d_in[1];  // (D, V)
    const float* J    = (const float*)d_in[2];  // (P, V, V)
    const float* bias = (const float*)d_in[3];  // (1,)
    // pair_i / pair_j (d_in[4], d_in[5]) are recomputed analytically.

    float* M = (float*)d_ws;                    // needs 1344*1344*4 = 7.2 MB

    const int total = NTOT * NTOT;
    build_M_kernel<<<(total + 255) / 256, 256, 0, stream>>>(J, M);

    const int nblocks = 32768 / BTILE;          // 2048 blocks of 16 batches
    fused_quadform_kernel<<<nblocks, 256, 0, stream>>>(x, h, M, bias, (float*)d_out);
}